// Angle_Feather_Extract_412316860911
// MI455X (gfx1250) — compile-verified
//
#include <hip/hip_runtime.h>

#define NB     2
#define NN     192
#define NFA    5
#define SPLIT  4
#define BLOCK  (SPLIT * NN)      // 768 threads = 24 wave32
#define CHUNKS (BLOCK / 4)       // 192 WMMA accumulation steps (4 partials each)

typedef __attribute__((ext_vector_type(2))) float v2f;
typedef __attribute__((ext_vector_type(8))) float v8f;

__launch_bounds__(BLOCK)
__global__ void fa_kernel(const float* __restrict__ dmat,
                          const float* __restrict__ cdm,
                          const float* __restrict__ fp,
                          const float* __restrict__ coeff,
                          float* __restrict__ out) {
  __shared__ float sRaw[NN];          // d[b,i,:]
  __shared__ float sD2[NN];           // d[b,i,:]^2
  __shared__ float sH[NN];            // 0.5/d (0 if d==0)
  __shared__ float sW[NN];            // cd0[b,i,:]
  __shared__ float P[BLOCK][16];      // per-thread partials, padded to 16 cols

  const int tid = threadIdx.x;
  const int k   = tid % NN;           // waves never straddle k (192 % 32 == 0)
  const int h   = tid / NN;           // j phase 0..3
  const int b   = blockIdx.x / NN;
  const int i   = blockIdx.x % NN;

  const float* drow_i = dmat + ((size_t)b * NN + i) * NN;
  const float* crow_i = cdm  + ((size_t)b * NN + i) * NN;

  // ---- async global->LDS staging of row i of d and cd (CDNA5 async path) ----
  if (h == 0) {
    unsigned lraw = (unsigned)(size_t)&sRaw[k];   // low 32 bits of flat LDS addr = LDS offset
    unsigned lw   = (unsigned)(size_t)&sW[k];
    const float* ga = drow_i + k;
    const float* gw = crow_i + k;
    asm volatile("global_load_async_to_lds_b32 %0, %1, off"
                 :: "v"(lraw), "v"(ga) : "memory");
    asm volatile("global_load_async_to_lds_b32 %0, %1, off"
                 :: "v"(lw), "v"(gw) : "memory");
    asm volatile("s_wait_asynccnt 0" ::: "memory");
  }
  __syncthreads();
  if (h == 0) {
    float v  = sRaw[k];
    sD2[k]   = v * v;
    sH[k]    = (v == 0.0f) ? 0.0f : 0.5f / v;     // exact 'where(den==0, 0, ...)' handling
    float w  = sW[k];
    sW[k]    = (w == 1.0f) ? 0.0f : w;            // cd0 = where(cd==1, 0, cd)
  }
  __syncthreads();

  // ---- per-thread constants ----
  float fpv[NFA], c2[NFA];
#pragma unroll
  for (int f = 0; f < NFA; ++f) {
    fpv[f] = fp[f];
    c2[f]  = -100.0f * 1.44269504088896340736f * coeff[f];  // exp(x) = exp2(log2e * x)
  }
  const float d2k  = sD2[k];
  const float invk = 2.0f * sH[k];                // 1/d[i,k] or 0
  const float wk   = sW[k];

  float acc[NFA] = {0.f, 0.f, 0.f, 0.f, 0.f};
  const float* dbase = dmat + (size_t)b * NN * NN;

  // ---- main loop: 48 j rows per thread, transcendental-bound ----
  for (int j = h; j < NN; j += SPLIT) {
    if (j + 4 * SPLIT < NN)
      __builtin_prefetch(&dbase[(size_t)(j + 4 * SPLIT) * NN + k], 0, 0);
    float djk = dbase[(size_t)j * NN + k];        // coalesced, L2-resident
    float num = sD2[j] + d2k - djk * djk;
    float a   = num * sH[j] * invk;               // 0 exactly on diagonal rows/cols
    float wjk = sW[j] * wk;
#pragma unroll
    for (int f = 0; f < NFA; ++f) {
      float diff = a - fpv[f];
      float t = __builtin_amdgcn_exp2f(c2[f] * (diff * diff));  // v_exp_f32
      acc[f] = fmaf(t, wjk, acc[f]);
    }
  }

  // ---- stash partials ----
#pragma unroll
  for (int f = 0; f < NFA; ++f) P[tid][f] = acc[f];
#pragma unroll
  for (int f = NFA; f < 16; ++f) P[tid][f] = 0.0f;
  __syncthreads();

  // ---- exact f32 reduction via V_WMMA_F32_16X16X4_F32 (wave 0 only) ----
  // A = all-ones 16x4  =>  D[m,n] = sum_{k=0..3} B[k,n] + C[m,n]
  // 4 independent accumulator chains break the WMMA->WMMA RAW dependency
  // (48 deep each instead of one 192-deep serial chain); combine with VALU.
  if (tid < 32) {
    const int lane = tid;
    const int n  = lane & 15;           // feature column (0..4 meaningful)
    const int kh = lane >> 4;           // which half of the K rows this lane feeds
    v8f c0 = {0.f, 0.f, 0.f, 0.f, 0.f, 0.f, 0.f, 0.f};
    v8f c1 = c0, c2a = c0, c3 = c0;
    v2f av; av.x = 1.0f; av.y = 1.0f;
    for (int cix = 0; cix < CHUNKS; cix += 4) {
      int base0 = (cix + 0) * 4 + kh * 2;
      int base1 = (cix + 1) * 4 + kh * 2;
      int base2 = (cix + 2) * 4 + kh * 2;
      int base3 = (cix + 3) * 4 + kh * 2;
      v2f b0; b0.x = P[base0 + 0][n]; b0.y = P[base0 + 1][n];
      v2f b1; b1.x = P[base1 + 0][n]; b1.y = P[base1 + 1][n];
      v2f b2; b2.x = P[base2 + 0][n]; b2.y = P[base2 + 1][n];
      v2f b3; b3.x = P[base3 + 0][n]; b3.y = P[base3 + 1][n];
      c0  = __builtin_amdgcn_wmma_f32_16x16x4_f32(false, av, false, b0, (short)0, c0,  false, false);
      c1  = __builtin_amdgcn_wmma_f32_16x16x4_f32(false, av, false, b1, (short)0, c1,  false, false);
      c2a = __builtin_amdgcn_wmma_f32_16x16x4_f32(false, av, false, b2, (short)0, c2a, false, false);
      c3  = __builtin_amdgcn_wmma_f32_16x16x4_f32(false, av, false, b3, (short)0, c3,  false, false);
    }
    if (lane < NFA) {
      float r = (c0[0] + c1[0]) + (c2a[0] + c3[0]);
      out[(size_t)blockIdx.x * NFA + lane] = r * (1.0f / (NN * 12.0f));
    }
  }
}

extern "C" void kernel_launch(void* const* d_in, const int* in_sizes, int n_in,
                              void* d_out, int out_size, void* d_ws, size_t ws_size,
                              hipStream_t stream) {
  const float* dmat  = (const float*)d_in[0];
  const float* cdm   = (const float*)d_in[1];
  const float* fp    = (const float*)d_in[2];
  const float* coeff = (const float*)d_in[3];
  float* out = (float*)d_out;
  dim3 grid(NB * NN);     // 384 workgroups, one per (b,i)
  dim3 block(BLOCK);      // 768 threads = 24 wave32
  hipLaunchKernelGGL(fa_kernel, grid, block, 0, stream, dmat, cdm, fp, coeff, out);
}